// OutAggregate_48120813584905
// MI455X (gfx1250) — compile-verified
//
#include <hip/hip_runtime.h>
#include <math.h>

// ---------------------------------------------------------------------------
// OutAggregate for MI455X (gfx1250, wave32).
// cross = lp @ p^T mapped onto V_WMMA_F32_16X16X4_F32 (f32 in/out, 16x16 tile
// per wave, K padded 91->96). Everything else is small, L2-resident work.
// ---------------------------------------------------------------------------

typedef float v2f __attribute__((ext_vector_type(2)));
typedef float v8f __attribute__((ext_vector_type(8)));

__device__ __forceinline__ float trunc4f(float x) {
    return truncf(x * 1.0e4f) / 1.0e4f;
}

// --- Kernel 1: boxes -> trunc4 -> cxcywh to xyxy ---------------------------
__global__ void k_prep_boxes(const float* __restrict__ bboxes,
                             float* __restrict__ xyxy, int N) {
    int i = blockIdx.x * blockDim.x + threadIdx.x;
    if (i >= N) return;
    float cx = trunc4f(bboxes[i * 4 + 0]);
    float cy = trunc4f(bboxes[i * 4 + 1]);
    float w  = trunc4f(bboxes[i * 4 + 2]);
    float h  = trunc4f(bboxes[i * 4 + 3]);
    xyxy[i * 4 + 0] = cx - 0.5f * w;
    xyxy[i * 4 + 1] = cy - 0.5f * h;
    xyxy[i * 4 + 2] = cx + 0.5f * w;
    xyxy[i * 4 + 3] = cy + 0.5f * h;
}

// --- Kernel 2: logits -> trunc4(sigmoid) -----------------------------------
__global__ void k_prep_logits(const float* __restrict__ logits,
                              float* __restrict__ xq, int total) {
    int e = blockIdx.x * blockDim.x + threadIdx.x;
    if (e >= total) return;
    float v = logits[e];
    float s = 1.0f / (1.0f + expf(-v));
    xq[e] = trunc4f(s);
}

// --- Kernel 3: pairwise GIoU -> bitmask (iou>0.9, no diag) + row counts ----
// One block per row i (mask is symmetric so cnt_row == cnt_col).
__global__ void k_giou_mask(const float* __restrict__ xyxy,
                            unsigned* __restrict__ mask,
                            unsigned* __restrict__ cnt, int N, int W) {
    const int i = blockIdx.x;
    const int tid = threadIdx.x;           // 128 threads = 4 waves
    const int lane = tid & 31;
    __shared__ unsigned s_cnt;
    if (tid == 0) s_cnt = 0u;
    __syncthreads();

    const float ax0 = xyxy[i * 4 + 0], ay0 = xyxy[i * 4 + 1];
    const float ax1 = xyxy[i * 4 + 2], ay1 = xyxy[i * 4 + 3];
    const float areaA = (ax1 - ax0) * (ay1 - ay0);

    unsigned local = 0u;
    for (int jb = 0; jb < N; jb += blockDim.x) {
        int j = jb + tid;
        float bx0 = xyxy[j * 4 + 0], by0 = xyxy[j * 4 + 1];
        float bx1 = xyxy[j * 4 + 2], by1 = xyxy[j * 4 + 3];
        float areaB = (bx1 - bx0) * (by1 - by0);
        float ix0 = fmaxf(ax0, bx0), iy0 = fmaxf(ay0, by0);
        float ix1 = fminf(ax1, bx1), iy1 = fminf(ay1, by1);
        float iw = fmaxf(ix1 - ix0, 0.0f), ih = fmaxf(iy1 - iy0, 0.0f);
        float inter = iw * ih;
        float uni = areaA + areaB - inter;
        float iou = inter / uni;
        float cx0 = fminf(ax0, bx0), cy0 = fminf(ay0, by0);
        float cx1 = fmaxf(ax1, bx1), cy1 = fmaxf(ay1, by1);
        float cw = fmaxf(cx1 - cx0, 0.0f), ch = fmaxf(cy1 - cy0, 0.0f);
        float areaC = cw * ch;
        float giou = iou - (areaC - uni) / areaC;
        bool pred = (giou > 0.9f) && (j != i);
        unsigned bal = __builtin_amdgcn_ballot_w32(pred);
        if (lane == 0) {
            mask[(size_t)i * W + (j >> 5)] = bal;
            local += (unsigned)__popc(bal);
        }
    }
    if (lane == 0 && local) atomicAdd(&s_cnt, local);
    __syncthreads();
    if (tid == 0) cnt[i] = s_cnt;
}

// --- Kernel 4: per-class count-weighted logsumexp --------------------------
// lse[c] = log( sum_i cnt[i] * exp(x[i,c]) ); x in [0,1], cnt<=N => no overflow.
__global__ void k_lse(const float* __restrict__ xq,
                      const unsigned* __restrict__ cnt,
                      float* __restrict__ lse, int N, int C) {
    const int c = blockIdx.x;
    __shared__ float sh[256];
    float acc = 0.0f;
    for (int i = threadIdx.x; i < N; i += blockDim.x)
        acc += (float)cnt[i] * expf(xq[(size_t)i * C + c]);
    sh[threadIdx.x] = acc;
    __syncthreads();
    for (int s = blockDim.x >> 1; s > 0; s >>= 1) {
        if (threadIdx.x < s) sh[threadIdx.x] += sh[threadIdx.x + s];
        __syncthreads();
    }
    if (threadIdx.x == 0) lse[c] = logf(sh[0]);
}

// --- Kernel 5: lp = x - lse, p = exp(lp) (K-padded to CP), negent[i] -------
__global__ void k_lp_p_negent(const float* __restrict__ xq,
                              const float* __restrict__ lse,
                              float* __restrict__ lp, float* __restrict__ p,
                              float* __restrict__ negent, int N, int C, int CP) {
    const int i = blockIdx.x;
    const int c = threadIdx.x;             // 128 threads
    __shared__ float sh[128];
    float contrib = 0.0f;
    if (c < CP) {
        float lv = 0.0f, pv = 0.0f;
        if (c < C) {
            lv = xq[(size_t)i * C + c] - lse[c];
            pv = expf(lv);
            contrib = pv * lv;
        }
        lp[(size_t)i * CP + c] = lv;
        p[(size_t)i * CP + c] = pv;
    }
    sh[c] = contrib;
    __syncthreads();
    for (int s = 64; s > 0; s >>= 1) {
        if (c < s) sh[c] += sh[c + s];
        __syncthreads();
    }
    if (c == 0) negent[i] = sh[0];
}

// --- Kernel 6: init agg bitmask with the identity (eye) --------------------
__global__ void k_init_agg(unsigned* __restrict__ agg, int N, int W) {
    int idx = blockIdx.x * blockDim.x + threadIdx.x;
    if (idx >= N * W) return;
    int i = idx / W, w = idx - i * W;
    agg[idx] = (w == (i >> 5)) ? (1u << (i & 31)) : 0u;
}

// --- Kernel 7: WMMA cross tile + KL threshold -> agg bits ------------------
// One wave per 16x16 tile of cross = lp @ p^T. Skips tiles whose IoU mask is
// empty (uniform ballot early-out keeps EXEC all-ones for the WMMAs).
__global__ void k_wmma_cross_agg(const float* __restrict__ lp,
                                 const float* __restrict__ p,
                                 const float* __restrict__ negent,
                                 const unsigned* __restrict__ mask,
                                 unsigned* __restrict__ agg,
                                 int N, int W, int CP, int C) {
    const int lane = threadIdx.x & 31;
    const int wave = threadIdx.x >> 5;
    const int j0 = (blockIdx.x * 4 + wave) * 16;
    const int i0 = blockIdx.y * 16;
    const int wj = j0 >> 5;
    const unsigned sh = (unsigned)(j0 & 31);   // 0 or 16

    // tile-empty early out (wave-uniform): any IoU>0.9 bits in this 16x16?
    {
        unsigned mw = mask[(size_t)(i0 + (lane & 15)) * W + wj];
        unsigned halfw = (mw >> sh) & 0xFFFFu;
        if (__builtin_amdgcn_ballot_w32(halfw != 0u) == 0u) return;
    }

    const int mA = lane & 15;                  // A/B row within tile
    const int kOff = (lane >> 4) * 2;          // K offset inside group-of-4
    const float nej = negent[j0 + (lane & 15)];

    const float* arow = lp + (size_t)(i0 + mA) * CP;
    const float* brow = p  + (size_t)(j0 + mA) * CP;

    v8f acc = {};
    for (int k0 = 0; k0 < CP; k0 += 4) {
        v2f a = *(const v2f*)(arow + k0 + kOff);
        v2f b = *(const v2f*)(brow + k0 + kOff);
        acc = __builtin_amdgcn_wmma_f32_16x16x4_f32(
            /*neg_a=*/false, a, /*neg_b=*/false, b,
            /*c_mod=*/(short)0, acc, /*reuse_a=*/false, /*reuse_b=*/false);
    }

    const float invC = 1.0f / (float)C;
    const int mBase = (lane >> 4) * 8;         // D layout: m = v + 8*(lane/16)
    const int n = lane & 15;                   //           n = lane % 16
#pragma unroll
    for (int v = 0; v < 8; ++v) {
        int i = i0 + mBase + v;
        unsigned mw = mask[(size_t)i * W + wj];
        bool mb = (mw >> (sh + (unsigned)n)) & 1u;
        float kl = (nej - acc[v]) * invC;
        bool bit = mb && (kl < 0.3f);
        unsigned bal = __builtin_amdgcn_ballot_w32(bit);
        if (lane == 0) {
            unsigned lo = bal & 0xFFFFu;       // rows i0+v      (lanes 0..15)
            unsigned hi = bal >> 16;           // rows i0+v+8    (lanes 16..31)
            if (lo) atomicOr(&agg[(size_t)(i0 + v) * W + wj], lo << sh);
            if (hi) atomicOr(&agg[(size_t)(i0 + 8 + v) * W + wj], hi << sh);
        }
    }
}

// --- Kernel 8: sparse row aggregation: out[i] = (agg[i] @ feat) / popcnt ----
__global__ void k_aggregate(const unsigned* __restrict__ agg,
                            const float* __restrict__ bboxes,
                            const float* __restrict__ logits,
                            float* __restrict__ out, int N, int C, int W) {
    const int i = blockIdx.x;
    const int f = threadIdx.x;                 // 0..3 -> bbox, 4..4+C-1 -> logit
    const int NF = 4 + C;
    if (f >= NF) return;
    const bool isBox = (f < 4);
    const float* src = isBox ? (bboxes + f) : (logits + (f - 4));
    const int stride = isBox ? 4 : C;

    float acc = 0.0f;
    int denom = 0;
    for (int w = 0; w < W; ++w) {
        unsigned word = agg[(size_t)i * W + w];
        if (!word) continue;
        denom += __popc(word);
        int jbase = w << 5;
        while (word) {
            int b = __ffs(word) - 1;
            acc += src[(size_t)(jbase + b) * stride];
            word &= word - 1u;
        }
    }
    float r = acc / (float)denom;              // denom >= 1 (diagonal)
    if (isBox) out[(size_t)i * 4 + f] = r;
    else       out[(size_t)N * 4 + (size_t)i * C + (f - 4)] = r;
}

// ---------------------------------------------------------------------------
extern "C" void kernel_launch(void* const* d_in, const int* in_sizes, int n_in,
                              void* d_out, int out_size, void* d_ws, size_t ws_size,
                              hipStream_t stream) {
    const float* bboxes = (const float*)d_in[0];
    const float* logits = (const float*)d_in[1];
    float* out = (float*)d_out;

    const int N = in_sizes[0] / 4;             // 4096
    const int C = in_sizes[1] / N;             // 91
    const int CP = (C + 15) & ~15;             // pad K to 96 (mult of 4, 16B rows)
    const int W = N / 32;                      // mask words per row

    // workspace carve-out (256B aligned slices)
    size_t off = 0;
    char* base = (char*)d_ws;
    auto carve = [&](size_t bytes) -> void* {
        void* pp = base + off;
        off += (bytes + 255) & ~(size_t)255;
        return pp;
    };
    float*    xyxy  = (float*)   carve((size_t)N * 4 * sizeof(float));
    float*    xq    = (float*)   carve((size_t)N * C * sizeof(float));
    unsigned* mask  = (unsigned*)carve((size_t)N * W * sizeof(unsigned));
    unsigned* cnt   = (unsigned*)carve((size_t)N * sizeof(unsigned));
    float*    lse   = (float*)   carve((size_t)CP * sizeof(float));
    float*    lpb   = (float*)   carve((size_t)N * CP * sizeof(float));
    float*    pb    = (float*)   carve((size_t)N * CP * sizeof(float));
    float*    ngent = (float*)   carve((size_t)N * sizeof(float));
    unsigned* agg   = (unsigned*)carve((size_t)N * W * sizeof(unsigned));
    (void)ws_size; (void)n_in; (void)out_size;

    k_prep_boxes<<<(N + 255) / 256, 256, 0, stream>>>(bboxes, xyxy, N);
    k_prep_logits<<<(N * C + 255) / 256, 256, 0, stream>>>(logits, xq, N * C);
    k_giou_mask<<<N, 128, 0, stream>>>(xyxy, mask, cnt, N, W);
    k_lse<<<C, 256, 0, stream>>>(xq, cnt, lse, N, C);
    k_lp_p_negent<<<N, 128, 0, stream>>>(xq, lse, lpb, pb, ngent, N, C, CP);
    k_init_agg<<<(N * W + 255) / 256, 256, 0, stream>>>(agg, N, W);
    dim3 grid(N / 64, N / 16);                 // 4 tiles (waves) per 128-thread block
    k_wmma_cross_agg<<<grid, 128, 0, stream>>>(lpb, pb, ngent, mask, agg, N, W, CP, C);
    k_aggregate<<<N, 128, 0, stream>>>(agg, bboxes, logits, out, N, C, W);
}